// ChamferDistanceLoss_42812234006906
// MI455X (gfx1250) — compile-verified
//
#include <hip/hip_runtime.h>

typedef __attribute__((ext_vector_type(2))) float v2f;
typedef __attribute__((ext_vector_type(8))) float v8f;

#define B_    4
#define HW_   76800          // 240*320 pixels per image
#define MBINS 256
#define NBLK  32             // wave-blocks per batch in kchamfer
#define TILES_TOTAL   (HW_ / 16)            // 4800 pixel tiles
#define TILES_PER_BLK (TILES_TOTAL / NBLK)  // 150
#define FINF_BITS 0x7F800000                // +inf as int bits

__device__ __forceinline__ int imin(int a, int b) { return a < b ? a : b; }

// ---------------- init: per-(batch,bin) min slots = +inf bits ----------------
__global__ void kinit(int* __restrict__ binmin) {
    int i = blockIdx.x * blockDim.x + threadIdx.x;
    if (i < B_ * MBINS) binmin[i] = FINF_BITS;
}

// ---------------- per-batch max reduction ----------------
__global__ void kmax(const float* __restrict__ target, float* __restrict__ maxv) {
    const int b = blockIdx.x;
    const float* p = target + b * HW_;
    float m = 0.0f;
    for (int i = threadIdx.x; i < HW_; i += blockDim.x) m = fmaxf(m, p[i]);
    __shared__ float s[256];
    s[threadIdx.x] = m;
    __syncthreads();
    for (int off = 128; off > 0; off >>= 1) {
        if ((int)threadIdx.x < off) s[threadIdx.x] = fmaxf(s[threadIdx.x], s[threadIdx.x + off]);
        __syncthreads();
    }
    if (threadIdx.x == 0) maxv[b] = s[0];
}

// ---------------- fused Chamfer pass: WMMA pairwise distance tiles ----------------
// sq[m,n] = bn[m]^2 - 2*bn[m]*tn[n] + tn[n]^2  ==  (16x4) x (4x16) GEMM
//   A row m = [bn^2, -2bn, 1, 0]    B col n = [1, tn, tn^2, 0]
// A layout (f32 16x4): lanes 0-15 hold M=0..15, v0=K0, v1=K1; lanes 16-31: v0=K2, v1=K3
// B layout (f32 4x16): v0 = rows K0 (lanes 0-15) | K2 (lanes 16-31); v1 = K1 | K3
// D layout: VGPR r, lanes 0-15 -> (m=r, n=lane); lanes 16-31 -> (m=r+8, n=lane-16)
// All mins are done on signed-int bit patterns: matches float ordering for the
// non-negative squared distances, avoids NaN-canonicalization VALU ops.
__global__ void __launch_bounds__(32)
kchamfer(const float* __restrict__ target, const float* __restrict__ bins,
         const float* __restrict__ maxv, int* __restrict__ binmin,
         float* __restrict__ t2b_partial) {
    const int lane = threadIdx.x;          // 0..31, one wave32 per block
    const int blk  = blockIdx.x;           // 0..NBLK-1
    const int b    = blockIdx.y;           // batch
    const int l15  = lane & 15;
    const bool lo  = lane < 16;

    const float inv = 1.0f / maxv[b];
    const float* tb = target + b * HW_;
    const float* bb = bins + b * MBINS;

    // A fragments for all 16 bin tiles (bins 16T..16T+15)
    v2f a[16];
#pragma unroll
    for (int T = 0; T < 16; ++T) {
        float bn = bb[T * 16 + l15] * inv;          // all lanes load (uniform halves)
        a[T].x = lo ? bn * bn      : 1.0f;          // K0 | K2
        a[T].y = lo ? -2.0f * bn   : 0.0f;          // K1 | K3
    }

    // running per-bin mins (int bit patterns) and t2b sum
    int dmin[16][8];
#pragma unroll
    for (int T = 0; T < 16; ++T)
#pragma unroll
        for (int r = 0; r < 8; ++r) dmin[T][r] = FINF_BITS;
    float sum = 0.0f;
    v8f cz = {};

    for (int k = 0; k < TILES_PER_BLK; ++k) {       // uniform trip count -> EXEC stays full
        const int tile = blk + k * NBLK;
        const float tn = tb[tile * 16 + l15] * inv;
        v2f bf;
        bf.x = lo ? 1.0f : tn * tn;                 // K0 | K2
        bf.y = lo ? tn   : 0.0f;                    // K1 | K3

        int pm = FINF_BITS;                         // per-pixel min over all 256 bins
#pragma unroll
        for (int T = 0; T < 16; ++T) {
            v8f d = __builtin_amdgcn_wmma_f32_16x16x4_f32(
                false, a[T], false, bf, (short)0, cz, false, false);
#pragma unroll
            for (int r = 0; r < 8; ++r) {
                const int di = __float_as_int(d[r]);
                dmin[T][r] = imin(dmin[T][r], di);  // v_min_i32, no canonicalize
                pm = imin(pm, di);
            }
        }
        // combine the two half-wave row groups (m=0..7 with m=8..15) for this pixel
        pm = imin(pm, __shfl_xor(pm, 16, 32));
        sum += lo ? __int_as_float(pm) : 0.0f;      // each pixel counted once
    }

    // per-bin min across the 16 pixel columns held in each half-wave
#pragma unroll
    for (int T = 0; T < 16; ++T) {
#pragma unroll
        for (int r = 0; r < 8; ++r) {
            int v = dmin[T][r];
            v = imin(v, __shfl_xor(v, 1, 32));
            v = imin(v, __shfl_xor(v, 2, 32));
            v = imin(v, __shfl_xor(v, 4, 32));
            v = imin(v, __shfl_xor(v, 8, 32));
            if (lane == 0)  atomicMin(&binmin[b * MBINS + T * 16 + r],     v);
            if (lane == 16) atomicMin(&binmin[b * MBINS + T * 16 + r + 8], v);
        }
    }

    // wave sum of per-pixel mins (lanes >= 16 contributed 0)
    sum += __shfl_xor(sum, 16, 32);
    sum += __shfl_xor(sum, 8, 32);
    sum += __shfl_xor(sum, 4, 32);
    sum += __shfl_xor(sum, 2, 32);
    sum += __shfl_xor(sum, 1, 32);
    if (lane == 0) t2b_partial[b * NBLK + blk] = sum;
}

// ---------------- finalize: fixed-order reduction -> scalar loss ----------------
__global__ void kfinal(const int* __restrict__ binmin,
                       const float* __restrict__ t2b_partial,
                       float* __restrict__ out) {
    __shared__ float s1[256], s2[256];
    const int tid = threadIdx.x;
    float a = 0.0f, c = 0.0f;
    for (int i = tid; i < B_ * MBINS; i += 256) a += __int_as_float(binmin[i]);
    for (int i = tid; i < B_ * NBLK;  i += 256) c += t2b_partial[i];
    s1[tid] = a; s2[tid] = c;
    __syncthreads();
    for (int off = 128; off > 0; off >>= 1) {
        if (tid < off) { s1[tid] += s1[tid + off]; s2[tid] += s2[tid + off]; }
        __syncthreads();
    }
    if (tid == 0)
        out[0] = (s1[0] / (float)MBINS + s2[0] / (float)HW_) / (float)B_ * 10.0f;
}

extern "C" void kernel_launch(void* const* d_in, const int* in_sizes, int n_in,
                              void* d_out, int out_size, void* d_ws, size_t ws_size,
                              hipStream_t stream) {
    const float* target = (const float*)d_in[0];   // [B,1,H,W] f32
    const float* bins   = (const float*)d_in[1];   // [B,M] f32

    float* ws     = (float*)d_ws;
    float* maxv   = ws;                             // 4 floats
    int*   binmin = (int*)(ws + 4);                 // B*M ints (float bit patterns)
    float* t2b    = ws + 4 + B_ * MBINS;            // B*NBLK floats

    kinit   <<<dim3(4),        dim3(256), 0, stream>>>(binmin);
    kmax    <<<dim3(B_),       dim3(256), 0, stream>>>(target, maxv);
    kchamfer<<<dim3(NBLK, B_), dim3(32),  0, stream>>>(target, bins, maxv, binmin, t2b);
    kfinal  <<<dim3(1),        dim3(256), 0, stream>>>(binmin, t2b, (float*)d_out);
}